// DNA_Encoder_SubGraph_59425167507618
// MI455X (gfx1250) — compile-verified
//
#include <hip/hip_runtime.h>
#include <hip/hip_bf16.h>
#include <math.h>

typedef __attribute__((ext_vector_type(2))) float v2f;
typedef __attribute__((ext_vector_type(8))) float v8f;

#define NNODES  10000
#define NEDGES  160000
#define INDIM   512
#define CDIM    128
#define NCLS    40
#define NLAYERS 4
#define NHEADS  8
#define DKH     16
#define KTOP    16
#define XSTRIDE (5*CDIM)   // x_all: [N, 5, 128]

#define NROWT   625        // N/16 row tiles
#define NCHUNK  25         // column chunks for sim sweep
#define CHUNKT  25         // column tiles per chunk (25*25 = 625)

// ---------------------------------------------------------------------------
// helpers
// ---------------------------------------------------------------------------
__global__ void zero_kernel(float* __restrict__ p, int n) {
  int i = blockIdx.x * blockDim.x + threadIdx.x;
  if (i < n) p[i] = 0.0f;
}

// deg = segment_sum(1, col) over E edges + N self loops
__global__ void deg_kernel(const int* __restrict__ col, float* __restrict__ deg) {
  int e = blockIdx.x * blockDim.x + threadIdx.x;
  if (e < NEDGES)                 atomicAdd(&deg[col[e]], 1.0f);
  else if (e < NEDGES + NNODES)   atomicAdd(&deg[e - NEDGES], 1.0f);
}

__global__ void dinv_kernel(const float* __restrict__ deg, float* __restrict__ dinv) {
  int i = blockIdx.x * blockDim.x + threadIdx.x;
  if (i < NNODES) dinv[i] = (deg[i] > 0.0f) ? rsqrtf(deg[i]) : 0.0f;
}

// ---------------------------------------------------------------------------
// GEMM1: out(slice0 of x_all) = relu(x[10000,512] @ W1[512,128] + b1)
// one wave per 16x16 tile, f32 WMMA 16x16x4, K-loop over 512
// ---------------------------------------------------------------------------
__global__ __launch_bounds__(32)
void gemm1_kernel(const float* __restrict__ x, const float* __restrict__ W1,
                  const float* __restrict__ b1, float* __restrict__ out) {
  int tile = blockIdx.x;          // 625*8 tiles
  int tj = tile & 7;              // column tile (C/16 = 8)
  int ti = tile >> 3;             // row tile    (N/16 = 625)
  int lane = threadIdx.x & 31;
  int l    = lane & 15;
  int koff = (lane >> 4) << 1;    // 0 or 2
  int m0 = ti * 16, n0 = tj * 16;

  v8f c = {};
  const float* arow = x + (size_t)(m0 + l) * INDIM + koff;
  for (int kb = 0; kb < INDIM; kb += 4) {
    v2f a, b;
    a.x = arow[kb];
    a.y = arow[kb + 1];
    b.x = W1[(size_t)(kb + koff) * CDIM + n0 + l];
    b.y = W1[(size_t)(kb + koff + 1) * CDIM + n0 + l];
    c = __builtin_amdgcn_wmma_f32_16x16x4_f32(false, a, false, b, (short)0, c,
                                              false, false);
  }
  int rbase = m0 + ((lane >> 4) << 3);
#pragma unroll
  for (int r = 0; r < 8; ++r) {
    float v = c[r] + b1[n0 + l];
    out[(size_t)(rbase + r) * XSTRIDE + n0 + l] = v > 0.0f ? v : 0.0f;
  }
}

// ---------------------------------------------------------------------------
// grouped linear: out[n,s,c] = sum_i xin[n,s,g*8+i] * W[g,i,o] + b[c]
// ---------------------------------------------------------------------------
__global__ void glin_kernel(const float* __restrict__ xin, int in_stride, int nslices,
                            const float* __restrict__ W, const float* __restrict__ bvec,
                            float* __restrict__ out) {
  int t = blockIdx.x * blockDim.x + threadIdx.x;
  int total = NNODES * nslices * CDIM;
  if (t >= total) return;
  int c = t % CDIM;
  int s = (t / CDIM) % nslices;
  int n = t / (CDIM * nslices);
  int g = c >> 3, o = c & 7;
  const float* xi = xin + (size_t)n * in_stride + (size_t)s * CDIM + g * 8;
  float acc = bvec[c];
#pragma unroll
  for (int i = 0; i < 8; ++i) acc += xi[i] * W[((size_t)g * 8 + i) * 8 + o];
  out[((size_t)n * nslices + s) * CDIM + c] = acc;
}

// ---------------------------------------------------------------------------
// edge attention: thread per (edge, head). edges = E real + N self loops.
// ---------------------------------------------------------------------------
__global__ void edge_attn_kernel(const int* __restrict__ ei, const float* __restrict__ dinv,
                                 const float* __restrict__ q, const float* __restrict__ k,
                                 const float* __restrict__ v, float* __restrict__ agg,
                                 int L) {
  int t = blockIdx.x * blockDim.x + threadIdx.x;
  int total = (NEDGES + NNODES) * NHEADS;
  if (t >= total) return;
  int h = t & (NHEADS - 1);
  int e = t >> 3;
  int r, cd;
  if (e < NEDGES) { r = ei[e]; cd = ei[NEDGES + e]; }
  else            { r = cd = e - NEDGES; }
  float nrm = dinv[r] * dinv[cd];

  const float* qp = q + (size_t)cd * CDIM + h * DKH;
  float qreg[DKH];
#pragma unroll
  for (int d = 0; d < DKH; ++d) qreg[d] = qp[d];

  float sc[NLAYERS];
  float mx = -1e30f;
  for (int s = 0; s < L; ++s) {
    const float* kp = k + ((size_t)r * L + s) * CDIM + h * DKH;
    float acc = 0.0f;
#pragma unroll
    for (int d = 0; d < DKH; ++d) acc += qreg[d] * kp[d];
    acc *= 0.25f;                     // 1/sqrt(DK)=1/4
    sc[s] = acc;
    if (acc > mx) mx = acc;
  }
  float ssum = 0.0f;
  for (int s = 0; s < L; ++s) { sc[s] = expf(sc[s] - mx); ssum += sc[s]; }
  float scale = nrm / ssum;

  float outv[DKH];
#pragma unroll
  for (int d = 0; d < DKH; ++d) outv[d] = 0.0f;
  for (int s = 0; s < L; ++s) {
    const float* vp = v + ((size_t)r * L + s) * CDIM + h * DKH;
    float a = sc[s];
#pragma unroll
    for (int d = 0; d < DKH; ++d) outv[d] += a * vp[d];
  }
  float* ap = agg + (size_t)cd * CDIM + h * DKH;
#pragma unroll
  for (int d = 0; d < DKH; ++d) atomicAdd(&ap[d], outv[d] * scale);
}

__global__ void relu_append_kernel(const float* __restrict__ agg,
                                   float* __restrict__ x_all, int slot) {
  int t = blockIdx.x * blockDim.x + threadIdx.x;
  if (t >= NNODES * CDIM) return;
  int n = t / CDIM, c = t % CDIM;
  float v = agg[t];
  x_all[(size_t)n * XSTRIDE + slot * CDIM + c] = v > 0.0f ? v : 0.0f;
}

// emb = x_all[:,4,:]; en = emb / max(||emb||, 1e-8); also emit emb to d_out
__global__ void norm_emb_kernel(const float* __restrict__ x_all,
                                float* __restrict__ en, float* __restrict__ emb_out) {
  int n = blockIdx.x * blockDim.x + threadIdx.x;
  if (n >= NNODES) return;
  const float* e = x_all + (size_t)n * XSTRIDE + 4 * CDIM;
  float ss = 0.0f;
  for (int c = 0; c < CDIM; ++c) { float v = e[c]; ss += v * v; }
  float nrm = sqrtf(ss);
  if (nrm < 1e-8f) nrm = 1e-8f;
  float inv = 1.0f / nrm;
  for (int c = 0; c < CDIM; ++c) {
    en[(size_t)n * CDIM + c]      = e[c] * inv;
    emb_out[(size_t)n * CDIM + c] = e[c];
  }
}

// ---------------------------------------------------------------------------
// async DMA of one contiguous 8KB B-tile (16 rows x 512B of en) into LDS:
// 16 x global_load_async_to_lds_b128, all 32 lanes moving 16B each.
// ---------------------------------------------------------------------------
__device__ __forceinline__ void async_tile_load(const float* gsrc, unsigned lbase,
                                                int lane) {
  unsigned long long g = (unsigned long long)(uintptr_t)gsrc;
#pragma unroll
  for (int i = 0; i < 16; ++i) {
    unsigned laddr = lbase + (unsigned)(i * 512 + lane * 16);
    unsigned long long gaddr = g + (unsigned long long)(i * 512 + lane * 16);
    asm volatile("global_load_async_to_lds_b128 %0, %1, off"
                 :: "v"(laddr), "v"(gaddr) : "memory");
  }
}

// ---------------------------------------------------------------------------
// sim = en @ en.T fused with per-(row,chunk) top-16 partials.
// grid (625 row-tiles, 25 chunks); one wave per block; A panel in registers;
// B tiles double-buffered in LDS via async loads; f32 WMMA 16x16x4.
// ---------------------------------------------------------------------------
__global__ __launch_bounds__(32)
void sim_topk_partial_kernel(const float* __restrict__ en,
                             float* __restrict__ ptkv, int* __restrict__ ptki) {
  __shared__ __align__(16) float bufB[2][16 * CDIM];   // 2 x 8KB
  __shared__ float tileC[16 * 16];
  int ti = blockIdx.x;            // row tile
  int ch = blockIdx.y;            // column chunk
  int lane = threadIdx.x & 31;
  int l    = lane & 15;
  int koff = (lane >> 4) << 1;
  int m0 = ti * 16;
  int t0 = ch * CHUNKT;           // first column tile of this chunk

  // preload A panel: 32 K-steps of v2f (16x128 block of en)
  v2f areg[32];
  const float* ap = en + (size_t)(m0 + l) * CDIM + koff;
#pragma unroll
  for (int kk = 0; kk < 32; ++kk) { areg[kk].x = ap[kk * 4]; areg[kk].y = ap[kk * 4 + 1]; }

  float vals[KTOP];
  int   idxs[KTOP];
#pragma unroll
  for (int t = 0; t < KTOP; ++t) { vals[t] = -1e30f; idxs[t] = 0x7fffffff; }

  unsigned lb[2];
  lb[0] = (unsigned)(uintptr_t)&bufB[0][0];
  lb[1] = (unsigned)(uintptr_t)&bufB[1][0];

  // prime pipeline with tile t0
  async_tile_load(en + (size_t)t0 * 16 * CDIM, lb[0], lane);

  for (int jj = 0; jj < CHUNKT; ++jj) {
    int tj = t0 + jj;
    int slot = jj & 1;
    if (jj + 1 < CHUNKT) {
      async_tile_load(en + (size_t)(tj + 1) * 16 * CDIM, lb[(jj + 1) & 1], lane);
      asm volatile("s_wait_asynccnt 0x10" ::: "memory");   // current tile landed
    } else {
      asm volatile("s_wait_asynccnt 0x0" ::: "memory");
    }
    __syncthreads();

    int n0 = tj * 16;
    const float* bp = &bufB[slot][l * CDIM + koff];
    v8f c = {};
#pragma unroll
    for (int kk = 0; kk < 32; ++kk) {
      v2f b; b.x = bp[kk * 4]; b.y = bp[kk * 4 + 1];
      c = __builtin_amdgcn_wmma_f32_16x16x4_f32(false, areg[kk], false, b, (short)0,
                                                c, false, false);
    }
    // dump C tile to LDS: VGPR r -> row (r + 8*(lane>=16)), col l
    int rbase = (lane >> 4) << 3;
#pragma unroll
    for (int r = 0; r < 8; ++r) tileC[(rbase + r) * 16 + l] = c[r];
    __syncthreads();
    if (lane < 16) {
#pragma unroll
      for (int j = 0; j < 16; ++j) {
        float v = tileC[lane * 16 + j];
        int id = n0 + j;
        if (v > vals[KTOP - 1]) {           // strict: earlier index wins ties
          vals[KTOP - 1] = v; idxs[KTOP - 1] = id;
#pragma unroll
          for (int t = KTOP - 1; t > 0; --t) {
            if (vals[t] > vals[t - 1]) {
              float tv = vals[t]; vals[t] = vals[t - 1]; vals[t - 1] = tv;
              int   td = idxs[t]; idxs[t] = idxs[t - 1]; idxs[t - 1] = td;
            }
          }
        }
      }
    }
    // all LDS reads of bufB[slot] must retire before it is re-targeted
    asm volatile("s_wait_dscnt 0x0" ::: "memory");
    __syncthreads();
  }
  if (lane < 16) {
#pragma unroll
    for (int t = 0; t < KTOP; ++t) {
      ptkv[((size_t)(m0 + lane) * NCHUNK + ch) * KTOP + t] = vals[t];
      ptki[((size_t)(m0 + lane) * NCHUNK + ch) * KTOP + t] = idxs[t];
    }
  }
}

// merge 25 per-chunk top-16 lists per row into the final top-16.
// exact lexicographic (value desc, index asc) => jax.lax.top_k semantics.
__global__ void topk_merge_kernel(const float* __restrict__ ptkv, const int* __restrict__ ptki,
                                  float* __restrict__ tkv, int* __restrict__ tki) {
  int n = blockIdx.x * blockDim.x + threadIdx.x;
  if (n >= NNODES) return;
  float vals[KTOP]; int idxs[KTOP];
#pragma unroll
  for (int t = 0; t < KTOP; ++t) { vals[t] = -1e30f; idxs[t] = 0x7fffffff; }
  for (int ch = 0; ch < NCHUNK; ++ch) {
    const float* pv = ptkv + ((size_t)n * NCHUNK + ch) * KTOP;
    const int*   pi = ptki + ((size_t)n * NCHUNK + ch) * KTOP;
    for (int t = 0; t < KTOP; ++t) {
      float v = pv[t]; int id = pi[t];
      bool beat = (v > vals[KTOP - 1]) ||
                  (v == vals[KTOP - 1] && id < idxs[KTOP - 1]);
      if (!beat) break;          // chunk lists are sorted: rest cannot beat tail
      vals[KTOP - 1] = v; idxs[KTOP - 1] = id;
#pragma unroll
      for (int s = KTOP - 1; s > 0; --s) {
        bool up = (vals[s] > vals[s - 1]) ||
                  (vals[s] == vals[s - 1] && idxs[s] < idxs[s - 1]);
        if (up) {
          float tv = vals[s]; vals[s] = vals[s - 1]; vals[s - 1] = tv;
          int   td = idxs[s]; idxs[s] = idxs[s - 1]; idxs[s - 1] = td;
        }
      }
    }
  }
#pragma unroll
  for (int t = 0; t < KTOP; ++t) {
    tkv[(size_t)n * KTOP + t] = vals[t];
    tki[(size_t)n * KTOP + t] = idxs[t];
  }
}

// ---------------------------------------------------------------------------
// classifier logits: thread per (n, c)
// ---------------------------------------------------------------------------
__global__ void logits_kernel(const float* __restrict__ emb, const float* __restrict__ W2,
                              const float* __restrict__ b2, float* __restrict__ logits) {
  int t = blockIdx.x * blockDim.x + threadIdx.x;
  if (t >= NNODES * NCLS) return;
  int c = t % NCLS, n = t / NCLS;
  const float* e = emb + (size_t)n * CDIM;
  float acc = b2[c];
  for (int i = 0; i < CDIM; ++i) acc += e[i] * W2[(size_t)i * NCLS + c];
  logits[t] = acc;
}

// final = 0.5*log_softmax(logits) + 0.5*log_softmax(sum_k exp(vals_k)*onehot(y[idx_k]))
__global__ void final_kernel(const float* __restrict__ logits, const float* __restrict__ tkv,
                             const int* __restrict__ tki, const int* __restrict__ y,
                             float* __restrict__ outp) {
  int n = blockIdx.x * blockDim.x + threadIdx.x;
  if (n >= NNODES) return;
  const float* lg = logits + (size_t)n * NCLS;
  float m1 = -1e30f;
  for (int c = 0; c < NCLS; ++c) m1 = fmaxf(m1, lg[c]);
  float s1 = 0.0f;
  for (int c = 0; c < NCLS; ++c) s1 += expf(lg[c] - m1);
  float ls1 = logf(s1);

  float ev[KTOP]; int lab[KTOP];
#pragma unroll
  for (int t = 0; t < KTOP; ++t) {
    ev[t]  = expf(tkv[(size_t)n * KTOP + t]);
    lab[t] = y[tki[(size_t)n * KTOP + t]];
  }
  float m2 = -1e30f;
  for (int c = 0; c < NCLS; ++c) {
    float sc = 0.0f;
#pragma unroll
    for (int t = 0; t < KTOP; ++t) sc += (lab[t] == c) ? ev[t] : 0.0f;
    m2 = fmaxf(m2, sc);
  }
  float s2 = 0.0f;
  for (int c = 0; c < NCLS; ++c) {
    float sc = 0.0f;
#pragma unroll
    for (int t = 0; t < KTOP; ++t) sc += (lab[t] == c) ? ev[t] : 0.0f;
    s2 += expf(sc - m2);
  }
  float ls2 = logf(s2);

  float* op = outp + (size_t)n * NCLS;
  for (int c = 0; c < NCLS; ++c) {
    float sc = 0.0f;
#pragma unroll
    for (int t = 0; t < KTOP; ++t) sc += (lab[t] == c) ? ev[t] : 0.0f;
    op[c] = 0.5f * (lg[c] - m1 - ls1) + 0.5f * (sc - m2 - ls2);
  }
}

// ---------------------------------------------------------------------------
// host-side orchestration
// ---------------------------------------------------------------------------
static inline int gdiv(int a, int b) { return (a + b - 1) / b; }

extern "C" void kernel_launch(void* const* d_in, const int* in_sizes, int n_in,
                              void* d_out, int out_size, void* d_ws, size_t ws_size,
                              hipStream_t stream) {
  (void)in_sizes; (void)n_in; (void)out_size; (void)ws_size;
  const float* x   = (const float*)d_in[0];
  const float* W1  = (const float*)d_in[1];
  const float* b1  = (const float*)d_in[2];
  const float* Wq  = (const float*)d_in[3];
  const float* bq  = (const float*)d_in[4];
  const float* Wk  = (const float*)d_in[5];
  const float* bk  = (const float*)d_in[6];
  const float* Wv  = (const float*)d_in[7];
  const float* bv  = (const float*)d_in[8];
  const float* W2  = (const float*)d_in[9];
  const float* b2  = (const float*)d_in[10];
  const int*   ei  = (const int*)d_in[11];   // [2, E]: row then col
  const int*   y   = (const int*)d_in[12];

  float* final_out = (float*)d_out;                      // [N, 40]
  float* emb_out   = final_out + (size_t)NNODES * NCLS;  // [N, 128]

  // workspace carve (floats)
  float* w      = (float*)d_ws;
  float* deg    = w;                       w += NNODES;
  float* dinv   = w;                       w += NNODES;
  float* x_all  = w;                       w += (size_t)NNODES * XSTRIDE;
  float* qbuf   = w;                       w += (size_t)NNODES * CDIM;
  float* kbuf   = w;                       w += (size_t)NNODES * NLAYERS * CDIM;
  float* vbuf   = w;                       w += (size_t)NNODES * NLAYERS * CDIM;
  float* agg    = w;                       w += (size_t)NNODES * CDIM;
  float* en     = w;                       w += (size_t)NNODES * CDIM;
  float* ptkv   = w;                       w += (size_t)NNODES * NCHUNK * KTOP;
  int*   ptki   = (int*)w;                 w += (size_t)NNODES * NCHUNK * KTOP;
  float* tkv    = w;                       w += (size_t)NNODES * KTOP;
  int*   tki    = (int*)w;                 w += (size_t)NNODES * KTOP;
  float* logits = w;                       w += (size_t)NNODES * NCLS;

  // gcn_norm
  zero_kernel<<<gdiv(NNODES, 256), 256, 0, stream>>>(deg, NNODES);
  deg_kernel<<<gdiv(NEDGES + NNODES, 256), 256, 0, stream>>>(ei + NEDGES, deg);
  dinv_kernel<<<gdiv(NNODES, 256), 256, 0, stream>>>(deg, dinv);

  // h = relu(x@W1+b1) -> x_all slice 0
  gemm1_kernel<<<NROWT * 8, 32, 0, stream>>>(x, W1, b1, x_all);

  for (int l = 0; l < NLAYERS; ++l) {
    int L = l + 1;
    glin_kernel<<<gdiv(NNODES * CDIM, 256), 256, 0, stream>>>(
        x_all + (size_t)l * CDIM, XSTRIDE, 1,
        Wq + (size_t)l * 1024, bq + (size_t)l * CDIM, qbuf);
    glin_kernel<<<gdiv(NNODES * L * CDIM, 256), 256, 0, stream>>>(
        x_all, XSTRIDE, L, Wk + (size_t)l * 1024, bk + (size_t)l * CDIM, kbuf);
    glin_kernel<<<gdiv(NNODES * L * CDIM, 256), 256, 0, stream>>>(
        x_all, XSTRIDE, L, Wv + (size_t)l * 1024, bv + (size_t)l * CDIM, vbuf);

    zero_kernel<<<gdiv(NNODES * CDIM, 256), 256, 0, stream>>>(agg, NNODES * CDIM);
    edge_attn_kernel<<<gdiv((NEDGES + NNODES) * NHEADS, 256), 256, 0, stream>>>(
        ei, dinv, qbuf, kbuf, vbuf, agg, L);
    relu_append_kernel<<<gdiv(NNODES * CDIM, 256), 256, 0, stream>>>(agg, x_all, l + 1);
  }

  norm_emb_kernel<<<gdiv(NNODES, 256), 256, 0, stream>>>(x_all, en, emb_out);
  sim_topk_partial_kernel<<<dim3(NROWT, NCHUNK), 32, 0, stream>>>(en, ptkv, ptki);
  topk_merge_kernel<<<gdiv(NNODES, 256), 256, 0, stream>>>(ptkv, ptki, tkv, tki);
  logits_kernel<<<gdiv(NNODES * NCLS, 256), 256, 0, stream>>>(emb_out, W2, b2, logits);
  final_kernel<<<gdiv(NNODES, 256), 256, 0, stream>>>(logits, tkv, tki, y, final_out);
}